// Conv2dSOT_70042326663754
// MI455X (gfx1250) — compile-verified
//
#include <hip/hip_runtime.h>

// ---------------------------------------------------------------------------
// SOM tree-descent (Conv2dSOT) for gfx1250 / MI455X.
//   Kernel 0 (som_init):     ||node_c||^2 into ws.
//   Kernel 1 (som_main):     S = X @ N^T via V_WMMA_F32_16X16X4_F32 (full fp32
//                            precision -> argmin identical to reference),
//                            tree descent on q = ||n||^2 - 2*dot, write BMU.
//   Kernel 2 (som_reduce):   owner-computes leaf reduction: 1 WG per leaf,
//                            scan L2-resident BMU ids, gather matching patches
//                            into private LDS bin (ds_add_f32, no global
//                            atomics), write M_leaf / C_leaf.
//   Kernel 3 (som_finalize): leaf prefix sums -> subtree sums -> node update,
//                            using lrates[k] = a0*2^k  =>
//                            lrs = a0*(1 + sum_t 2^(t-1)*[ancestor match]).
// Roofline: mandatory traffic ~2x16MB image + 32MB L2-hit id scans; no atomic
// hot spots -> ~1.5 us class, GEMM (2.1 GFLOP fp32 WMMA) fully hidden.
// ---------------------------------------------------------------------------

typedef float v2f __attribute__((ext_vector_type(2)));
typedef float v4f __attribute__((ext_vector_type(4)));
typedef float v8f __attribute__((ext_vector_type(8)));

#define IMG_W     2048
#define NPATCH    65536      // 256 x 256 patches
#define NUM_NODES 255
#define DEPTH     7
#define NODE_OUT  (NUM_NODES * 64)   // 16320 floats (first output)
#define A0        0.00234375f        // 0.3 * 2 / 256

// workspace layout (floats)
#define WS_M   0        // 128 leaves x 64 dims
#define WS_C   8192     // 128 leaf counts
#define WS_NN  8320     // 256 node norms (255 used)

#define WAVES  2        // waves per workgroup in som_main
#define SPAD   264      // padded score row (256 cols + pad, dodge bank conflicts)

__global__ __launch_bounds__(256) void som_init(const float* __restrict__ nodes,
                                                float* __restrict__ ws) {
    int tid = threadIdx.x;
    float s = 0.0f;
    if (tid < NUM_NODES) {
        for (int k = 0; k < 64; ++k) { float v = nodes[tid * 64 + k]; s += v * v; }
    }
    if (tid < 256) ws[WS_NN + tid] = s;
}

__global__ __launch_bounds__(WAVES * 32) void som_main(const float* __restrict__ img,
                                                       const float* __restrict__ nodes,
                                                       const float* __restrict__ ws,
                                                       int* __restrict__ out_bmu) {
    __shared__ float S[WAVES * 16 * SPAD];

    const int tid  = threadIdx.x;
    const int w    = tid >> 5;
    const int lane = tid & 31;
    const int half = lane >> 4;     // half-wave = K split for f32 WMMA frags
    const int lm   = lane & 15;

    const int patchBase = (blockIdx.x * WAVES + w) * 16;
    const int p  = patchBase + lm;
    const int pr = p >> 8, pc = p & 255;
    const float* prow = img + (size_t)pr * 8 * IMG_W + pc * 8;

    // A fragments: 16x4 f32 per WMMA. Lane (lm, half) holds A[lm][4kk+2h+{0,1}].
    v2f a[16];
#pragma unroll
    for (int kk = 0; kk < 16; ++kk) {
        int k = kk * 4 + half * 2;                  // element index in 8x8 patch
        a[kk] = *(const v2f*)(prow + (k >> 3) * IMG_W + (k & 7));
    }

    const float* nn = ws + WS_NN;
    float* Sw = S + w * 16 * SPAD;

#pragma unroll 1
    for (int t = 0; t < 16; ++t) {                  // 16 column tiles -> 256 cols
        const int col  = t * 16 + lm;
        const int ccol = (col < NUM_NODES) ? col : 0;   // clamp pad col (no OOB)
        v8f acc = {};
#pragma unroll
        for (int kk = 0; kk < 16; ++kk) {
            int k = kk * 4 + half * 2;
            v2f b = *(const v2f*)(nodes + ccol * 64 + k);   // B[k][n]=nodes[col][k]
            acc = __builtin_amdgcn_wmma_f32_16x16x4_f32(
                      false, a[kk], false, b, (short)0, acc, false, false);
        }
        // adjusted score q = ||n||^2 - 2*dot (same argmin as reference dist^2^2)
        float nv = nn[col];
#pragma unroll
        for (int i = 0; i < 8; ++i) {
            Sw[(i + 8 * half) * SPAD + col] = nv - 2.0f * acc[i];
        }
    }
    __syncthreads();

    // Tree descent: lanes 0..15 handle one patch each (one row of Sw).
    if (lane < 16) {
        const float* q = Sw + lane * SPAD;
        int cur = 0;
        for (int l = 0; l < DEPTH; ++l) {
            int c0 = 2 * cur + 1, c1 = c0 + 1;
            cur = (q[c1] < q[c0]) ? c1 : c0;
        }
        out_bmu[patchBase + lane] = cur;            // final BMU (leaf node index)
    }
}

// Owner-computes leaf reduction: one workgroup per leaf, zero global atomics.
__global__ __launch_bounds__(256) void som_reduce(const float* __restrict__ img,
                                                  const int* __restrict__ bmu,
                                                  float* __restrict__ ws) {
    __shared__ float bin[64];
    __shared__ float cnt;
    const int tid = threadIdx.x;
    const int g   = blockIdx.x;                     // leaf id 0..127
    if (tid < 64) bin[tid] = 0.0f;
    if (tid == 64) cnt = 0.0f;
    __syncthreads();

    const int target = 127 + g;
    int mycnt = 0;
    for (int p = tid; p < NPATCH; p += 256) {       // BMU array is L2-resident
        if (bmu[p] == target) {
            ++mycnt;
            const float* pim = img + (size_t)(p >> 8) * 8 * IMG_W + (p & 255) * 8;
#pragma unroll
            for (int r = 0; r < 8; ++r) {
                v4f x0 = *(const v4f*)(pim + r * IMG_W);
                v4f x1 = *(const v4f*)(pim + r * IMG_W + 4);
                atomicAdd(&bin[r * 8 + 0], x0.x);
                atomicAdd(&bin[r * 8 + 1], x0.y);
                atomicAdd(&bin[r * 8 + 2], x0.z);
                atomicAdd(&bin[r * 8 + 3], x0.w);
                atomicAdd(&bin[r * 8 + 4], x1.x);
                atomicAdd(&bin[r * 8 + 5], x1.y);
                atomicAdd(&bin[r * 8 + 6], x1.z);
                atomicAdd(&bin[r * 8 + 7], x1.w);
            }
        }
    }
    if (mycnt) atomicAdd(&cnt, (float)mycnt);       // LDS atomic
    __syncthreads();
    if (tid < 64) ws[WS_M + g * 64 + tid] = bin[tid];
    if (tid == 64) ws[WS_C + g] = cnt;
}

__global__ __launch_bounds__(256) void som_finalize(const float* __restrict__ nodes,
                                                    const float* __restrict__ ws,
                                                    float* __restrict__ out_nodes) {
    __shared__ float PrefM[129 * 64];   // exclusive-prefix over 128 leaves, per dim
    __shared__ float PrefC[129];
    const int tid = threadIdx.x;

    if (tid < 64) {
        float run = 0.0f;
        PrefM[tid] = 0.0f;
        for (int l = 0; l < 128; ++l) {
            run += ws[WS_M + l * 64 + tid];
            PrefM[(l + 1) * 64 + tid] = run;
        }
    } else if (tid == 64) {
        float run = 0.0f;
        PrefC[0] = 0.0f;
        for (int l = 0; l < 128; ++l) {
            run += ws[WS_C + l];
            PrefC[l + 1] = run;
        }
    }
    __syncthreads();

    const float invP = 1.0f / 65536.0f;
    for (int idx = tid; idx < 254 * 64; idx += 256) {
        int j = 1 + (idx >> 6);                     // node 1..254
        int k = idx & 63;
        int lvl = 31 - __clz(j + 1);
        float msum = 0.0f, csum = 0.0f;
        int v = j;
        for (int t = lvl; t >= 1; --t) {            // walk ancestors up to root
            int s  = 1 << (DEPTH - t);              // subtree leaf count of anc_t
            int lo = (v + 1) * s - 1 - 127;         // first leaf id of subtree
            float wgt = (float)(1 << (t - 1));      // lrates[t]-lrates[t-1] = a0*2^(t-1)
            msum += wgt * (PrefM[(lo + s) * 64 + k] - PrefM[lo * 64 + k]);
            csum += wgt * (PrefC[lo + s] - PrefC[lo]);
            v = (v - 1) >> 1;
        }
        float sall  = PrefM[128 * 64 + k];          // sum over all patches, dim k
        float nodev = nodes[j * 64 + k];
        float upd = A0 * (sall + msum) * invP
                  - A0 * (65536.0f + csum) * invP * nodev;
        out_nodes[j * 64 + k] = nodev + upd;
    }
    if (tid < 64) out_nodes[tid] = nodes[tid];      // root row unchanged
}

extern "C" void kernel_launch(void* const* d_in, const int* in_sizes, int n_in,
                              void* d_out, int out_size, void* d_ws, size_t ws_size,
                              hipStream_t stream) {
    (void)in_sizes; (void)n_in; (void)out_size; (void)ws_size;
    const float* img   = (const float*)d_in[0];
    const float* nodes = (const float*)d_in[1];
    float* out   = (float*)d_out;
    int*   bmu   = (int*)d_out + NODE_OUT;
    float* ws    = (float*)d_ws;

    som_init<<<1, 256, 0, stream>>>(nodes, ws);
    som_main<<<NPATCH / (WAVES * 16), WAVES * 32, 0, stream>>>(img, nodes, ws, bmu);
    som_reduce<<<128, 256, 0, stream>>>(img, bmu, ws);
    som_finalize<<<1, 256, 0, stream>>>(nodes, ws, out);
}